// Downsample_9010841387722
// MI455X (gfx1250) — compile-verified
//
#include <hip/hip_runtime.h>

typedef float v2f __attribute__((ext_vector_type(2)));
typedef float v8f __attribute__((ext_vector_type(8)));
typedef unsigned int tdm_v4u __attribute__((ext_vector_type(4)));
typedef int          tdm_v8i __attribute__((ext_vector_type(8)));
typedef int          tdm_v4i __attribute__((ext_vector_type(4)));

#define N_IN    1000000
#define M_OUT   125000
#define CIN     96
#define COUT    192
#define KCH     8
#define MTILE   64             // rows per block
#define LROW    108            // LDS row stride in DWORDs (96 data + TDM pad 4dw/32dw)
#define NWAVES  12             // 12 waves * 16 cols = 192 = COUT
#define BLOCKT  384
#define NCHUNK  128
#define ROWS_PER_CHUNK 977     // 977 * 128 = 125056 >= 125000

// ---------------------------------------------------------------------------
// TDM gather: one tensor op pulls 8 gathered rows of 96 f32 from `base` into
// LDS at lds_addr.  D# gather mode, 32-bit indices, pad 4 DWORDs every 32
// DWORDs so LDS row stride becomes 108 (conflict-free for the WMMA A reads).
// Negative / OOB indices (>= tensor_dim1 = N_IN) read back as zeros.
// ---------------------------------------------------------------------------
__device__ __forceinline__ void tdm_gather8(const float* base, const int* idx,
                                            unsigned int lds_addr) {
    unsigned long long ga = (unsigned long long)(uintptr_t)base;
    tdm_v4u g0;
    g0.x = 0xC0000001u;   // gather_mode=1 | gather_index_size=32b | count=1
    g0.y = (unsigned int)__builtin_amdgcn_readfirstlane((int)lds_addr);
    g0.z = (unsigned int)__builtin_amdgcn_readfirstlane((int)(unsigned int)ga);
    g0.w = (unsigned int)__builtin_amdgcn_readfirstlane((int)(unsigned int)(ga >> 32))
           | 0x80000000u; // type=2 ("image") in bits 127:126
    tdm_v8i g1;
    g1[0] = 0x07120000;   // pad_amount=4dw | pad_interval=32dw | pad_enable | data_size=4B
    g1[1] = 0x00600000;   // tensor_dim0 = 96 (bits 63:48)
    g1[2] = 0x42400000;   // tensor_dim1 lo16 (1000000 = 0xF4240) at bits 95:80
    g1[3] = 0x0060000F;   // tile_dim0 = 96 (bits 127:112) | tensor_dim1 hi bits
    g1[4] = 8;            // tile_dim1 = #gather indices
    g1[5] = 96;           // tensor_dim0_stride
    g1[6] = 0;
    g1[7] = 0;
    tdm_v4i g2, g3;
#pragma unroll
    for (int i = 0; i < 4; ++i) {
        g2[i] = __builtin_amdgcn_readfirstlane(idx[i]);
        g3[i] = __builtin_amdgcn_readfirstlane(idx[4 + i]);
    }
    tdm_v8i gx = {0, 0, 0, 0, 0, 0, 0, 0};   // extra operand (clang-23 6-arg form)
    __builtin_amdgcn_tensor_load_to_lds(g0, g1, g2, g3, gx, 0);
}

// ---------------------------------------------------------------------------
// Kernel 1: TDM-gather + GEMM via V_WMMA_F32_16X16X4_F32.
// Block: 64 output rows x 192 channels.  12 waves, each owns a 16-wide N tile
// and 4 M sub-tiles (B fragment reused 4x).  K processed child-by-child
// (8 chunks of 96), double-buffered in LDS, gather done by the TDM engine.
// ---------------------------------------------------------------------------
__global__ __launch_bounds__(BLOCKT) void octconv_wmma(
    const float* __restrict__ data,    // [N_IN, 96]
    const float* __restrict__ W,       // [768, 192]
    const float* __restrict__ bias,    // [192]
    const int*   __restrict__ neigh,   // [M_OUT, 8]
    float* __restrict__ out)           // [M_OUT, 192]
{
    __shared__ float Abuf[2][MTILE][LROW];   // 2 * 64 * 108 * 4 = 55296 B

    const int tid  = threadIdx.x;
    const int m0   = blockIdx.x * MTILE;
    const int lane = tid & 31;
    const int wave = tid >> 5;

    const unsigned int lds_base = (unsigned int)(uintptr_t)(void*)&Abuf[0][0][0];
    const unsigned int buf_bytes = (unsigned int)(MTILE * LROW * 4);

    // ---- issue helper: gather chunk (child) j into buffer `sel` -----------
    auto issue_chunk = [&](int j, int sel) {
        const unsigned int bbase = lds_base + (unsigned int)sel * buf_bytes;
#pragma unroll 1
        for (int opr = 0; opr < MTILE / 8; ++opr) {       // 8 TDM ops / chunk
            const int base_m = m0 + opr * 8;
            int idx[8];
#pragma unroll
            for (int i = 0; i < 8; ++i) {
                const int gm = base_m + i;
                idx[i] = (gm < M_OUT) ? neigh[gm * KCH + j] : -1;  // -1 -> OOB -> zeros
            }
            tdm_gather8(data, idx, bbase + (unsigned int)(opr * 8 * LROW * 4));
        }
    };

    // ---- per-lane WMMA fragment coordinates -------------------------------
    const int n    = wave * 16 + (lane & 15);  // output channel
    const int rowA = lane & 15;                // A fragment row within sub-tile
    const int koff = (lane >> 4) << 1;         // lanes 16..31 hold K+2,K+3

    v8f acc0, acc1, acc2, acc3;
    const float bv = bias[n];
#pragma unroll
    for (int i = 0; i < 8; ++i) { acc0[i] = bv; acc1[i] = bv; acc2[i] = bv; acc3[i] = bv; }

    // ---- software pipeline: TDM gather (wave 0) overlapped with WMMA ------
    if (wave == 0) issue_chunk(0, 0);

    for (int j = 0; j < KCH; ++j) {
        if (wave == 0) {
            if (j < KCH - 1) {
                issue_chunk(j + 1, (j + 1) & 1);
                __builtin_amdgcn_s_wait_tensorcnt(8);   // chunk j's 8 ops done
            } else {
                __builtin_amdgcn_s_wait_tensorcnt(0);
            }
        }
        __syncthreads();                                // chunk j visible to all

        const float* buf  = &Abuf[j & 1][0][0];
        const float* wrow = W + j * CIN * COUT + n;
#pragma unroll 4
        for (int kk0 = 0; kk0 < CIN; kk0 += 4) {
            const int kk  = kk0 + koff;
            const int off = kk + ((kk >> 5) << 2);      // +4 dw pad per 32 dw
            v2f b;
            b.x = wrow[kk * COUT];
            b.y = wrow[(kk + 1) * COUT];
            v2f a0 = *(const v2f*)(buf + (rowA)*LROW + off);
            v2f a1 = *(const v2f*)(buf + (16 + rowA) * LROW + off);
            v2f a2 = *(const v2f*)(buf + (32 + rowA) * LROW + off);
            v2f a3 = *(const v2f*)(buf + (48 + rowA) * LROW + off);
            acc0 = __builtin_amdgcn_wmma_f32_16x16x4_f32(false, a0, false, b, (short)0, acc0, false, false);
            acc1 = __builtin_amdgcn_wmma_f32_16x16x4_f32(false, a1, false, b, (short)0, acc1, false, false);
            acc2 = __builtin_amdgcn_wmma_f32_16x16x4_f32(false, a2, false, b, (short)0, acc2, false, false);
            acc3 = __builtin_amdgcn_wmma_f32_16x16x4_f32(false, a3, false, b, (short)0, acc3, false, false);
        }
        __syncthreads();                                // buf[j&1] free for j+2
    }

    // ---- store: C/D layout -> row = m0 + s*16 + 8*(lane>=16) + r ----------
    const int mhalf = (lane >> 4) << 3;
#pragma unroll
    for (int s = 0; s < 4; ++s) {
        const v8f acc = (s == 0) ? acc0 : (s == 1) ? acc1 : (s == 2) ? acc2 : acc3;
        const int mbase = m0 + s * 16 + mhalf;
#pragma unroll
        for (int r = 0; r < 8; ++r) {
            const int gm = mbase + r;
            if (gm < M_OUT) out[(long long)gm * COUT + n] = acc[r];
        }
    }
}

// ---------------------------------------------------------------------------
// Kernel 2: per-chunk partial sums / sums-of-squares (deterministic, coalesced)
// ---------------------------------------------------------------------------
__global__ __launch_bounds__(COUT) void bn_partial(
    const float* __restrict__ out, float* __restrict__ psum,
    float* __restrict__ psq)
{
    const int c  = threadIdx.x;
    const int b  = blockIdx.x;
    const int r0 = b * ROWS_PER_CHUNK;
    const int r1 = min(M_OUT, r0 + ROWS_PER_CHUNK);
    float s = 0.f, ss = 0.f;
    for (int r = r0; r < r1; ++r) {
        const float x = out[(long long)r * COUT + c];
        s += x;
        ss += x * x;
    }
    psum[b * COUT + c] = s;
    psq[b * COUT + c]  = ss;
}

// ---------------------------------------------------------------------------
// Kernel 3: reduce partials -> per-channel scale/shift
// ---------------------------------------------------------------------------
__global__ __launch_bounds__(COUT) void bn_finalize(
    const float* __restrict__ psum, const float* __restrict__ psq,
    const float* __restrict__ gamma, const float* __restrict__ beta,
    float* __restrict__ scale, float* __restrict__ shift)
{
    const int c = threadIdx.x;
    float s = 0.f, ss = 0.f;
    for (int b = 0; b < NCHUNK; ++b) {
        s  += psum[b * COUT + c];
        ss += psq[b * COUT + c];
    }
    const float inv_m = 1.0f / (float)M_OUT;
    const float mean  = s * inv_m;
    const float var   = ss * inv_m - mean * mean;
    const float rstd  = rsqrtf(var + 1e-5f);
    const float sc    = gamma[c] * rstd;
    scale[c] = sc;
    shift[c] = beta[c] - mean * sc;
}

// ---------------------------------------------------------------------------
// Kernel 4: in-place normalize (float4; 192 % 4 == 0 so channels never wrap)
// ---------------------------------------------------------------------------
__global__ __launch_bounds__(256) void bn_apply(
    float* __restrict__ out, const float* __restrict__ scale,
    const float* __restrict__ shift, int n4)
{
    const int i = blockIdx.x * blockDim.x + threadIdx.x;
    if (i >= n4) return;
    float4 x = ((float4*)out)[i];
    const int c0 = (i * 4) % COUT;
    const float4 sc = *(const float4*)(scale + c0);
    const float4 sh = *(const float4*)(shift + c0);
    x.x = x.x * sc.x + sh.x;
    x.y = x.y * sc.y + sh.y;
    x.z = x.z * sc.z + sh.z;
    x.w = x.w * sc.w + sh.w;
    ((float4*)out)[i] = x;
}

// ---------------------------------------------------------------------------
extern "C" void kernel_launch(void* const* d_in, const int* in_sizes, int n_in,
                              void* d_out, int out_size, void* d_ws, size_t ws_size,
                              hipStream_t stream) {
    // setup_inputs() dict order: data, weights, bias, gamma, beta, neigh
    const float* data  = (const float*)d_in[0];
    const float* W     = (const float*)d_in[1];
    const float* bias  = (const float*)d_in[2];
    const float* gamma = (const float*)d_in[3];
    const float* beta  = (const float*)d_in[4];
    const int*   neigh = (const int*)d_in[5];
    float* out = (float*)d_out;

    float* ws    = (float*)d_ws;
    float* psum  = ws;                        // [128][192]
    float* psq   = ws + NCHUNK * COUT;        // [128][192]
    float* scale = ws + 2 * NCHUNK * COUT;    // [192]
    float* shift = scale + COUT;              // [192]

    const int mtiles = (M_OUT + MTILE - 1) / MTILE;   // 1954
    octconv_wmma<<<mtiles, BLOCKT, 0, stream>>>(data, W, bias, neigh, out);

    bn_partial<<<NCHUNK, COUT, 0, stream>>>(out, psum, psq);
    bn_finalize<<<1, COUT, 0, stream>>>(psum, psq, gamma, beta, scale, shift);

    const int n4 = M_OUT * COUT / 4;
    bn_apply<<<(n4 + 255) / 256, 256, 0, stream>>>(out, scale, shift, n4);
}